// Model_14121852470238
// MI455X (gfx1250) — compile-verified
//
#include <hip/hip_runtime.h>
#include <math.h>
#include <stdint.h>

// ---------------------------------------------------------------------------
// Problem constants (fixed by the harness reference)
// ---------------------------------------------------------------------------
#define BS    8
#define TGT   256
#define SRC   512
#define HID   1024
#define TVOC  15000
#define VOC   20000

typedef __attribute__((ext_vector_type(2))) float v2f;
typedef __attribute__((ext_vector_type(8))) float v8f;

__device__ __constant__ float EPSF = 1.1920928955078125e-07f;  // np.float32 eps = 2^-23
#define LEPS   (-15.942385152878742f)                          // log(2^-23)
#define NEGBIG (-1.0e9f)

__device__ __forceinline__ float lse2(float a, float b) {
  float m = fmaxf(a, b);
  return m + logf(expf(a - m) + expf(b - m));
}

__device__ __forceinline__ v2f mkv2(float2 v) { v2f r; r.x = v.x; r.y = v.y; return r; }

// ---------------------------------------------------------------------------
// K1: query = gelu(feature @ W_q^T + b_q)   (2048x1024)
// One wave computes a 16x64 tile: 4 accumulators share one A fragment per
// k-step -> 4x less A traffic, 4 back-to-back v_wmma_f32_16x16x4_f32.
// ---------------------------------------------------------------------------
__global__ void k1_query_gemm(const float* __restrict__ feature,
                              const float* __restrict__ Wq,
                              const float* __restrict__ bq,
                              float* __restrict__ query) {
  const int lane  = threadIdx.x & 31;
  const int wv    = threadIdx.x >> 5;
  const int mrow  = lane & 15;          // A: M index / B: N index held by this lane
  const int khalf = (lane >> 4) << 1;   // lanes 0-15 -> K+0/K+1, lanes 16-31 -> K+2/K+3
  const int m0 = blockIdx.x * 16;
  const int n0 = (blockIdx.y * 8 + wv) * 64;

  const float* ap = feature + (size_t)(m0 + mrow) * HID + khalf;
  const float* bp = Wq      + (size_t)(n0 + mrow) * HID + khalf;

  v8f acc[4] = {{}, {}, {}, {}};
  for (int k = 0; k < HID; k += 4) {
    if ((k & 63) == 0) {                       // pull next cachelines early
      __builtin_prefetch(ap + k + 64, 0, 1);   // -> global_prefetch_b8
      __builtin_prefetch(bp + k + 64, 0, 1);
      __builtin_prefetch(bp + 16 * HID + k + 64, 0, 1);
      __builtin_prefetch(bp + 32 * HID + k + 64, 0, 1);
      __builtin_prefetch(bp + 48 * HID + k + 64, 0, 1);
    }
    const v2f a  = mkv2(*(const float2*)(ap + k));
    const v2f b0 = mkv2(*(const float2*)(bp + k));
    const v2f b1 = mkv2(*(const float2*)(bp + 16 * HID + k));
    const v2f b2 = mkv2(*(const float2*)(bp + 32 * HID + k));
    const v2f b3 = mkv2(*(const float2*)(bp + 48 * HID + k));
    acc[0] = __builtin_amdgcn_wmma_f32_16x16x4_f32(false, a, false, b0, (short)0, acc[0], false, false);
    acc[1] = __builtin_amdgcn_wmma_f32_16x16x4_f32(false, a, false, b1, (short)0, acc[1], false, false);
    acc[2] = __builtin_amdgcn_wmma_f32_16x16x4_f32(false, a, false, b2, (short)0, acc[2], false, false);
    acc[3] = __builtin_amdgcn_wmma_f32_16x16x4_f32(false, a, false, b3, (short)0, acc[3], false, false);
  }

  // C/D layout: lanes 0-15: M=r, N=lane ; lanes 16-31: M=8+r, N=lane-16
  const int mo = (lane >> 4) * 8;
  const int nl = lane & 15;
#pragma unroll
  for (int q4 = 0; q4 < 4; ++q4) {
    const int n = n0 + q4 * 16 + nl;
    const float bias = bq[n];
#pragma unroll
    for (int r = 0; r < 8; ++r) {
      const int m = m0 + mo + r;
      float x = acc[q4][r] + bias;
      float g = 0.5f * x * (1.0f + erff(x * 0.70710678118654752f)); // exact gelu
      query[(size_t)m * HID + n] = g;
    }
  }
}

// ---------------------------------------------------------------------------
// K2: scores[b,t,s] = dot(query[b,t,:], memory[b,s,:]) / sqrt(1024),  s < 512
// Same 16x64-per-wave register blocking.
// ---------------------------------------------------------------------------
__global__ void k2_score_gemm(const float* __restrict__ query,
                              const float* __restrict__ memory,
                              float* __restrict__ scores) {
  const int lane  = threadIdx.x & 31;
  const int wv    = threadIdx.x >> 5;
  const int mrow  = lane & 15;
  const int khalf = (lane >> 4) << 1;
  const int b  = blockIdx.z;
  const int m0 = blockIdx.x * 16;
  const int n0 = wv * 64;               // 8 waves x 64 = 512 = SRC

  const float* ap = query  + ((size_t)b * TGT + m0 + mrow) * HID + khalf;
  const float* bp = memory + ((size_t)b * SRC + n0 + mrow) * HID + khalf;

  v8f acc[4] = {{}, {}, {}, {}};
  for (int k = 0; k < HID; k += 4) {
    if ((k & 63) == 0) {
      __builtin_prefetch(ap + k + 64, 0, 1);
      __builtin_prefetch(bp + k + 64, 0, 1);
      __builtin_prefetch(bp + 16 * HID + k + 64, 0, 1);
      __builtin_prefetch(bp + 32 * HID + k + 64, 0, 1);
      __builtin_prefetch(bp + 48 * HID + k + 64, 0, 1);
    }
    const v2f a  = mkv2(*(const float2*)(ap + k));
    const v2f b0 = mkv2(*(const float2*)(bp + k));
    const v2f b1 = mkv2(*(const float2*)(bp + 16 * HID + k));
    const v2f b2 = mkv2(*(const float2*)(bp + 32 * HID + k));
    const v2f b3 = mkv2(*(const float2*)(bp + 48 * HID + k));
    acc[0] = __builtin_amdgcn_wmma_f32_16x16x4_f32(false, a, false, b0, (short)0, acc[0], false, false);
    acc[1] = __builtin_amdgcn_wmma_f32_16x16x4_f32(false, a, false, b1, (short)0, acc[1], false, false);
    acc[2] = __builtin_amdgcn_wmma_f32_16x16x4_f32(false, a, false, b2, (short)0, acc[2], false, false);
    acc[3] = __builtin_amdgcn_wmma_f32_16x16x4_f32(false, a, false, b3, (short)0, acc[3], false, false);
  }

  const int mo = (lane >> 4) * 8;
  const int nl = lane & 15;
#pragma unroll
  for (int q4 = 0; q4 < 4; ++q4) {
    const int n = n0 + q4 * 16 + nl;
#pragma unroll
    for (int r = 0; r < 8; ++r) {
      const int m = m0 + mo + r;
      scores[((size_t)b * TGT + m) * SRC + n] = acc[q4][r] * 0.03125f; // 1/sqrt(1024)
    }
  }
}

// ---------------------------------------------------------------------------
// K3: per (b,t): sentinel score + mask + log_softmax over 513.
//     Writes probs[s] = exp(log_softmax[s]) in-place over scores, and gate[bt].
// ---------------------------------------------------------------------------
__global__ void k3_softmax(float* __restrict__ scores,   // in: scores, out: probs
                           const float* __restrict__ query,
                           const float* __restrict__ sentinel,
                           const unsigned char* __restrict__ mask,
                           float* __restrict__ gate) {
  const int bt  = blockIdx.x;
  const int b   = bt >> 8;              // TGT == 256
  const int tid = threadIdx.x;
  __shared__ float srow[SRC + 1];
  __shared__ float red[256];

  // sentinel dot product (1024 MACs, block-parallel)
  const float* q = query + (size_t)bt * HID;
  float acc = 0.f;
  for (int h = tid; h < HID; h += 256) acc += q[h] * sentinel[h];
  red[tid] = acc; __syncthreads();
  for (int off = 128; off; off >>= 1) {
    if (tid < off) red[tid] += red[tid + off];
    __syncthreads();
  }
  const float sent = red[0] * 0.03125f;
  __syncthreads();

  float* srow_g = scores + (size_t)bt * SRC;
  for (int s = tid; s < SRC; s += 256) {
    float x = srow_g[s];
    if (mask[b * SRC + s]) x = NEGBIG;
    srow[s] = x;
  }
  if (tid == 0) srow[SRC] = sent;       // sentinel never masked
  __syncthreads();

  float m = -INFINITY;
  for (int i = tid; i <= SRC; i += 256) m = fmaxf(m, srow[i]);
  red[tid] = m; __syncthreads();
  for (int off = 128; off; off >>= 1) {
    if (tid < off) red[tid] = fmaxf(red[tid], red[tid + off]);
    __syncthreads();
  }
  m = red[0]; __syncthreads();

  float s_ = 0.f;
  for (int i = tid; i <= SRC; i += 256) s_ += expf(srow[i] - m);
  red[tid] = s_; __syncthreads();
  for (int off = 128; off; off >>= 1) {
    if (tid < off) red[tid] += red[tid + off];
    __syncthreads();
  }
  const float lse = m + logf(red[0]);

  if (tid == 0) gate[bt] = sent - lse;  // log-softmax of sentinel column
  for (int s = tid; s < SRC; s += 256) srow_g[s] = expf(srow[s] - lse);
}

// ---------------------------------------------------------------------------
// K4: fused logits log-softmax + default output.
//     The whole 15000-float logits row is staged ONCE into LDS using CDNA5
//     async direct-to-LDS copies (GLOBAL_LOAD_ASYNC_TO_LDS_B128, ASYNCcnt),
//     avoiding the global->VGPR->ds_store round trip.
//     out[bt,v<15000]  = logsumexp(logits_v - lse + g, log(EPS))
//     out[bt,v>=15000] = log(EPS)
// ---------------------------------------------------------------------------
__global__ void k4_rows(const float* __restrict__ logits,
                        const float* __restrict__ gate,
                        float* __restrict__ lse_out,
                        float* __restrict__ out) {
  const int bt  = blockIdx.x;
  const int tid = threadIdx.x;
  __shared__ __align__(16) float row[TVOC];
  __shared__ float red[256];

  // ---- async global -> LDS staging (512 B per wave per instruction) ----
  const float* lrow = logits + (size_t)bt * TVOC;
  const unsigned lds0 = (unsigned)(uintptr_t)(&row[0]);   // low 32 bits = LDS offset
  for (int i = tid; i < TVOC / 4; i += 256) {
    const float* g = lrow + 4 * i;
    const unsigned loff = lds0 + 16u * (unsigned)i;
    asm volatile("global_load_async_to_lds_b128 %0, %1, off"
                 :: "v"(loff), "v"(g) : "memory");
  }
  asm volatile("s_wait_asynccnt 0" ::: "memory");
  __syncthreads();

  float m = -INFINITY;
  for (int i = tid; i < TVOC; i += 256) m = fmaxf(m, row[i]);
  red[tid] = m; __syncthreads();
  for (int off = 128; off; off >>= 1) {
    if (tid < off) red[tid] = fmaxf(red[tid], red[tid + off]);
    __syncthreads();
  }
  m = red[0]; __syncthreads();

  float s_ = 0.f;
  for (int i = tid; i < TVOC; i += 256) s_ += expf(row[i] - m);
  red[tid] = s_; __syncthreads();
  for (int off = 128; off; off >>= 1) {
    if (tid < off) red[tid] += red[tid + off];
    __syncthreads();
  }
  const float lse = m + logf(red[0]);
  if (tid == 0) lse_out[bt] = lse;
  const float g = gate[bt];

  float4* o4 = (float4*)(out + (size_t)bt * VOC);
  const float4* r4 = (const float4*)row;
  const float sh = g - lse;
  for (int i = tid; i < TVOC / 4; i += 256) {
    float4 x = r4[i], y;
    y.x = lse2(x.x + sh, LEPS);
    y.y = lse2(x.y + sh, LEPS);
    y.z = lse2(x.z + sh, LEPS);
    y.w = lse2(x.w + sh, LEPS);
    o4[i] = y;
  }
  const float4 c = make_float4(LEPS, LEPS, LEPS, LEPS);
  for (int i = tid; i < (VOC - TVOC) / 4; i += 256) o4[TVOC / 4 + i] = c;
}

// ---------------------------------------------------------------------------
// K5: sparse pointer scatter. One block per (b,s). Only the first occurrence
//     of each vocab id in content_e[b,:] writes; it sums probs over all
//     duplicate source positions (faithful one_hot einsum merge).
// ---------------------------------------------------------------------------
__global__ void k5_scatter(const float* __restrict__ logits,
                           const int* __restrict__ content,
                           const float* __restrict__ probs,
                           const float* __restrict__ gate,
                           const float* __restrict__ lse,
                           float* __restrict__ out) {
  const int b   = blockIdx.x >> 9;      // SRC == 512
  const int s   = blockIdx.x & 511;
  const int tid = threadIdx.x;
  __shared__ int c_row[SRC];
  __shared__ int list[SRC];
  __shared__ int cnt;
  __shared__ int minrep;

  for (int i = tid; i < SRC; i += 256) c_row[i] = content[b * SRC + i];
  if (tid == 0) { cnt = 0; minrep = 0x7fffffff; }
  __syncthreads();

  const int v = c_row[s];
  for (int i = tid; i < SRC; i += 256) {
    if (c_row[i] == v) {
      atomicMin(&minrep, i);
      int p = atomicAdd(&cnt, 1);
      list[p] = i;
    }
  }
  __syncthreads();
  if (minrep != s) return;              // uniform across block

  const int n  = cnt;
  const int t  = tid;                   // 256 threads == TGT
  const size_t bt = (size_t)b * TGT + t;

  float p = 0.f;
  for (int j = 0; j < n; ++j) p += probs[bt * SRC + list[j]];
  const float branch2 = logf(p + EPSF); // (-L, clamp, +L) cancels exactly

  float r;
  if (v < TVOC) {
    const float a = logits[bt * TVOC + v] - lse[bt] + gate[bt];
    r = lse2(a, branch2);
  } else {
    r = branch2;                        // logsumexp(-inf, x) = x
  }
  out[bt * VOC + v] = r;
}

// ---------------------------------------------------------------------------
// Launch
// ---------------------------------------------------------------------------
extern "C" void kernel_launch(void* const* d_in, const int* in_sizes, int n_in,
                              void* d_out, int out_size, void* d_ws, size_t ws_size,
                              hipStream_t stream) {
  (void)in_sizes; (void)n_in; (void)out_size; (void)ws_size;
  const float* logits   = (const float*)d_in[0];
  const float* feature  = (const float*)d_in[1];
  const float* memory   = (const float*)d_in[2];
  const float* Wq       = (const float*)d_in[3];
  const float* bq       = (const float*)d_in[4];
  const float* sentinel = (const float*)d_in[5];
  const unsigned char* mask = (const unsigned char*)d_in[6];
  const int* content    = (const int*)d_in[7];
  float* out = (float*)d_out;

  // workspace layout (floats): query | scores/probs | gate | lse  (~12.6 MB)
  float* query  = (float*)d_ws;
  float* scores = query  + (size_t)BS * TGT * HID;
  float* gate   = scores + (size_t)BS * TGT * SRC;
  float* lse    = gate   + (size_t)BS * TGT;

  dim3 blk(256);
  // K1: M tiles = 2048/16 = 128 ; N: 2 blocks x 8 waves x 64 = 1024
  k1_query_gemm<<<dim3((BS * TGT) / 16, HID / 512), blk, 0, stream>>>(feature, Wq, bq, query);
  // K2: M tiles = 256/16 = 16 ; N: 8 waves x 64 = 512 ; z = batch
  k2_score_gemm<<<dim3(TGT / 16, 1, BS), blk, 0, stream>>>(query, memory, scores);
  k3_softmax<<<dim3(BS * TGT), blk, 0, stream>>>(scores, query, sentinel, mask, gate);
  k4_rows<<<dim3(BS * TGT), blk, 0, stream>>>(logits, gate, lse, out);
  k5_scatter<<<dim3(BS * SRC), blk, 0, stream>>>(logits, content, scores, gate, lse, out);
}